// NMS_v1_12335146074351
// MI455X (gfx1250) — compile-verified
//
#include <hip/hip_runtime.h>
#include <stdint.h>

#define BATCH 8
#define NPRED 200000
#define NCLS  81
#define TOPK  200

// ---------------- Kernel A: scores = max(softmax(logits)) ------------------
// 128 rows/block, tile = 128*81*4 = 41472 B (16B aligned chunks since
// 128*324 is a multiple of 16).  Async DMA global->LDS (CDNA5 path),
// then one row per lane reduced out of LDS.
#define ROWS_PB   128
#define THREADS_A 128

__global__ __launch_bounds__(THREADS_A)
void scores_kernel(const float* __restrict__ logits, float* __restrict__ scores)
{
    __shared__ __align__(16) float tile[ROWS_PB * NCLS];   // 41472 bytes
    const int tid  = threadIdx.x;
    const int row0 = blockIdx.x * ROWS_PB;

    const uint64_t gbase = (uint64_t)(uintptr_t)(logits + (size_t)row0 * NCLS);
    const uint32_t lbase = (uint32_t)(uintptr_t)&tile[0];  // low 32b = LDS offset

    // 2592 x 16-byte chunks, coalesced async copies tracked by ASYNCcnt.
    for (int i = tid; i < (ROWS_PB * NCLS) / 4; i += THREADS_A) {
        uint32_t goff  = (uint32_t)(i * 16);
        uint32_t laddr = lbase + (uint32_t)(i * 16);
        asm volatile("global_load_async_to_lds_b128 %0, %1, %2"
                     :: "v"(laddr), "v"(goff), "s"(gbase)
                     : "memory");
    }
    asm volatile("s_wait_asynccnt 0x0" ::: "memory");
    __syncthreads();

    // lane t reduces row t: stride 81 dwords between lanes -> 64-bank clean.
    const float* r = &tile[tid * NCLS];
    float m = r[0];
    #pragma unroll
    for (int c = 1; c < NCLS; ++c) m = fmaxf(m, r[c]);
    float s = 0.0f;
    #pragma unroll
    for (int c = 0; c < NCLS; ++c) s += __expf(r[c] - m);
    scores[row0 + tid] = 1.0f / s;     // max(softmax) = 1/sum(exp(x-max))
}

// ---------------- Kernel B: per-batch top-200 + greedy NMS -----------------
#define THREADS_B 1024
#define CHUNK     196          // 1024*196 = 200704 >= NPRED
#define NCHUNK    1024

__global__ __launch_bounds__(THREADS_B)
void topk_nms_kernel(const float* __restrict__ segs,
                     float* __restrict__ scores_ws,
                     int* __restrict__ out)
{
    const int b = blockIdx.x;
    const int t = threadIdx.x;
    float*       sc  = scores_ws + (size_t)b * NPRED;
    const float* seg = segs      + (size_t)b * NPRED * 2;

    __shared__ float cmax[NCHUNK];
    __shared__ int   cidx[NCHUNK];
    __shared__ float rv[THREADS_B];
    __shared__ int   ri[THREADS_B];
    __shared__ float top_val[TOPK];
    __shared__ int   top_idx[TOPK];
    __shared__ float sx1[TOPK], sx2[TOPK];
    __shared__ int   active[TOPK];
    __shared__ int   kept[TOPK];
    __shared__ int   win_e;

    const float NEG = -__builtin_inff();

    // ---- build per-chunk maxima (ties -> larger index, argsort-tail-like)
    {
        int s0 = t * CHUNK;
        int s1 = min(s0 + CHUNK, NPRED);
        float v = NEG; int ix = 0;
        for (int i = s0; i < s1; ++i) {
            float x = sc[i];
            if (x >= v) { v = x; ix = i; }
        }
        cmax[t] = v; cidx[t] = ix;
    }
    __syncthreads();

    // ---- 200 selection rounds: tree-reduce 1024 chunk maxima, then one
    //      wave rebuilds only the winning chunk's max.
    for (int k = 0; k < TOPK; ++k) {
        rv[t] = cmax[t]; ri[t] = cidx[t];
        __syncthreads();
        for (int off = THREADS_B / 2; off > 0; off >>= 1) {
            if (t < off) {
                float v2 = rv[t + off]; int i2 = ri[t + off];
                if (v2 > rv[t] || (v2 == rv[t] && i2 > ri[t])) { rv[t] = v2; ri[t] = i2; }
            }
            __syncthreads();
        }
        if (t == 0) {
            int e = ri[0];
            top_idx[k] = e; top_val[k] = rv[0];
            sc[e] = NEG;               // suppress in scratch copy
            win_e = e;
        }
        __syncthreads();
        int c = win_e / CHUNK;
        if (t < 32) {                  // wave 0 rescans the winning chunk
            int s0 = c * CHUNK;
            int s1 = min(s0 + CHUNK, NPRED);
            float lv = NEG; int lix = 0;
            for (int i = s0 + t; i < s1; i += 32) {
                float x = sc[i];
                if (x > lv || (x == lv && i > lix)) { lv = x; lix = i; }
            }
            for (int off = 16; off > 0; off >>= 1) {
                float ov = __shfl_down(lv, off);
                int   oi = __shfl_down(lix, off);
                if (ov > lv || (ov == lv && oi > lix)) { lv = ov; lix = oi; }
            }
            if (t == 0) { cmax[c] = lv; cidx[c] = lix; }
        }
        __syncthreads();
    }

    // ---- gather segments of the 200 candidates
    if (t < TOPK) {
        int g = top_idx[t];
        sx1[t] = seg[2 * g + 0];
        sx2[t] = seg[2 * g + 1];
        active[t] = 1;
    }
    __syncthreads();

    // ---- greedy NMS, 200 rounds (argmax among active, suppress overlaps)
    for (int r = 0; r < TOPK; ++r) {
        rv[t] = (t < TOPK && active[t]) ? top_val[t] : NEG;
        ri[t] = t;
        __syncthreads();
        for (int off = THREADS_B / 2; off > 0; off >>= 1) {
            if (t < off) {
                float v2 = rv[t + off]; int i2 = ri[t + off];
                if (v2 > rv[t] || (v2 == rv[t] && i2 < ri[t])) { rv[t] = v2; ri[t] = i2; }
            }
            __syncthreads();
        }
        int  j   = ri[0];
        bool any = rv[0] > NEG;
        __syncthreads();
        if (any) {
            if (t < TOPK && active[t]) {
                float raw = fminf(sx2[t], sx2[j]) - fmaxf(sx1[t], sx1[j]);
                if (raw > 0.0f) active[t] = 0;    // inter <= OVERLAP(0) keeps
            }
            if (t == j) active[t] = 0;
        }
        if (t == 0) kept[r] = any ? top_idx[j] : 0;
        __syncthreads();
    }

    if (t < TOPK) {
        out[b * TOPK + t] = kept[t];
        out[(BATCH + b) * TOPK + t] = 0;   // concatenated zeros half
    }
}

// --------------------------------- launch ---------------------------------
extern "C" void kernel_launch(void* const* d_in, const int* in_sizes, int n_in,
                              void* d_out, int out_size, void* d_ws, size_t ws_size,
                              hipStream_t stream)
{
    const float* logits = (const float*)d_in[0];   // (8, 200000, 81) f32
    const float* segs   = (const float*)d_in[1];   // (8, 200000, 2)  f32
    float* scores = (float*)d_ws;                  // 8*200000 f32 = 6.4 MB scratch
    int*   out    = (int*)d_out;                   // (16, 200) int32

    scores_kernel<<<(BATCH * NPRED) / ROWS_PB, THREADS_A, 0, stream>>>(logits, scores);
    topk_nms_kernel<<<BATCH, THREADS_B, 0, stream>>>(segs, scores, out);
}